// AttnDecoder_72653666779204
// MI455X (gfx1250) — compile-verified
//
#include <hip/hip_runtime.h>
#include <hip/hip_bf16.h>
#include <math.h>

// Sizes from the reference
#define BB 256      // batch
#define OO 1024     // output dim
#define HH 2048     // 2*hidden
#define SS 12       // encoder steps
#define KCAT (OO + HH)  // 3072

typedef __attribute__((ext_vector_type(2))) float v2f;
typedef __attribute__((ext_vector_type(8))) float v8f;

// ---------------------------------------------------------------------------
// WMMA fp32 GEMM: out[M,N] = act( X[M,K] @ W[N,K]^T + bias0 + bias1 [+ out] )
// Tile: workgroup = 8 waves = 256 threads, owns 64(M) x 128(N).
// Each wave: 16 N columns, 4 M-subtiles of 16 -> 4 x v8f accumulators.
// X chunk [64 x 128] staged in LDS (row stride 132 -> bank-conflict-free
// ds_load_b64 A-fragment reads). W streamed from global: per K-step of 4,
// each lane loads exactly its half of the 4x16 B tile as one b64.
// flags: bit0 = relu epilogue, bit1 = accumulate previous out
// ---------------------------------------------------------------------------
#define KC 128
#define XS_LD 132   // 128 + 4 pad: lane l banks = 4*l (+col) -> no conflicts

__global__ __launch_bounds__(256) void wmma_gemm_kernel(
    const float* __restrict__ X, const float* __restrict__ W,
    const float* __restrict__ bias0, const float* __restrict__ bias1,
    float* __restrict__ out, int M, int N, int K, int flags)
{
    __shared__ __align__(16) float Xs[64 * XS_LD];

    const int n_block = blockIdx.x * 128;
    const int m_block = blockIdx.y * 64;
    const int tid  = threadIdx.x;
    const int wave = tid >> 5;
    const int lane = tid & 31;
    const int half = lane >> 4;     // 0: K{0,1}, 1: K{2,3} (A/B VGPR layout)
    const int l16  = lane & 15;
    const int n_lane = n_block + wave * 16 + l16;

    v8f acc[4] = {};

    // Per-lane streaming base into W: row n_lane, offset by this lane-half's K pair
    const float* __restrict__ Wrow = W + (size_t)n_lane * K + 2 * half;

    for (int k0 = 0; k0 < K; k0 += KC) {
        __syncthreads();
        // Cooperative stage of X[m_block..+63][k0..k0+127]: 32 floats/thread
        for (int i = tid; i < 64 * (KC / 4); i += 256) {
            const int row = i >> 5;          // KC/4 = 32 float4 per row
            const int c4  = i & 31;
            const float4 v = *(const float4*)(X + (size_t)(m_block + row) * K + k0 + c4 * 4);
            *(float4*)(&Xs[row * XS_LD + c4 * 4]) = v;
        }
        __syncthreads();

        // Prefetch next W chunk for this lane's stream (global_prefetch_b8)
        if (k0 + KC < K) __builtin_prefetch(Wrow + k0 + KC, 0, 0);

        for (int kk = 0; kk < KC; kk += 4) {
            // B fragment (4x16 K x N tile): lane holds W[n][k + 2*half .. +1]
            const v2f b = *(const v2f*)(Wrow + k0 + kk);
            #pragma unroll
            for (int t = 0; t < 4; ++t) {
                // A fragment (16x4): lanes 0-15 -> K{kk,kk+1}, 16-31 -> K{kk+2,kk+3}
                const v2f a = *(const v2f*)(&Xs[(t * 16 + l16) * XS_LD + kk + 2 * half]);
                acc[t] = __builtin_amdgcn_wmma_f32_16x16x4_f32(
                    false, a, false, b, (short)0, acc[t], false, false);
            }
        }
    }

    // Epilogue. C layout: VGPR r, lanes 0-15 -> M=r, lanes 16-31 -> M=8+r; N=l16
    float bsum = 0.0f;
    if (bias0) bsum += bias0[n_lane];
    if (bias1) bsum += bias1[n_lane];
    const bool relu = (flags & 1) != 0;
    const bool accp = (flags & 2) != 0;
    #pragma unroll
    for (int t = 0; t < 4; ++t) {
        #pragma unroll
        for (int r = 0; r < 8; ++r) {
            const int m = m_block + t * 16 + half * 8 + r;
            const size_t idx = (size_t)m * N + n_lane;
            float v = acc[t][r] + bsum;
            if (accp) v += out[idx];
            if (relu) v = fmaxf(v, 0.0f);
            out[idx] = v;
        }
    }
}

// ---------------------------------------------------------------------------
// Attention: scores over S=12 encoder steps, softmax, context; emits
// xc[b] = concat(x0[b], attn_applied[b])  -> [B, 3072]
// One workgroup per batch row.
// ---------------------------------------------------------------------------
__global__ __launch_bounds__(256) void attn_kernel(
    const float* __restrict__ x,       // [1,B,O]
    const float* __restrict__ h0,      // [2,B,H2] (layer 0 at offset 0)
    const float* __restrict__ enc,     // [S,B,H2]
    const float* __restrict__ attn_W,  // [S, 3072]
    const float* __restrict__ attn_b,  // [S]
    float* __restrict__ xc)            // [B, 3072]
{
    const int b = blockIdx.x;
    const int tid = threadIdx.x;
    __shared__ float red[256];
    __shared__ float wsm[SS];

    const float* __restrict__ x0b = x + (size_t)b * OO;
    const float* __restrict__ h0b = h0 + (size_t)b * HH;

    for (int s = 0; s < SS; ++s) {
        float p = 0.0f;
        const float* __restrict__ wr = attn_W + (size_t)s * KCAT;
        for (int k = tid; k < KCAT; k += 256) {
            const float a = (k < OO) ? x0b[k] : h0b[k - OO];
            p += a * wr[k];
        }
        red[tid] = p;
        __syncthreads();
        for (int off = 128; off > 0; off >>= 1) {
            if (tid < off) red[tid] += red[tid + off];
            __syncthreads();
        }
        if (tid == 0) wsm[s] = red[0] + attn_b[s];
        __syncthreads();
    }

    if (tid == 0) {
        float mx = wsm[0];
        for (int s = 1; s < SS; ++s) mx = fmaxf(mx, wsm[s]);
        float sum = 0.0f;
        for (int s = 0; s < SS; ++s) { const float e = expf(wsm[s] - mx); wsm[s] = e; sum += e; }
        const float inv = 1.0f / sum;
        for (int s = 0; s < SS; ++s) wsm[s] *= inv;
    }
    __syncthreads();

    float* __restrict__ xcb = xc + (size_t)b * KCAT;
    for (int j = tid; j < OO; j += 256) xcb[j] = x0b[j];
    for (int h = tid; h < HH; h += 256) {
        float a = 0.0f;
        #pragma unroll
        for (int s = 0; s < SS; ++s)
            a += wsm[s] * enc[((size_t)s * BB + b) * HH + h];
        xcb[OO + h] = a;
    }
}

// ---------------------------------------------------------------------------
// LSTM pointwise cell: gates [B, 4*H2] in (i,f,g,o) order
// ---------------------------------------------------------------------------
__device__ __forceinline__ float sigf(float v) { return 1.0f / (1.0f + expf(-v)); }

__global__ __launch_bounds__(256) void lstm_cell_kernel(
    const float* __restrict__ gates, const float* __restrict__ c_prev,
    float* __restrict__ h_out, float* __restrict__ c_out)
{
    const int idx = blockIdx.x * 256 + threadIdx.x;   // over B*H2
    if (idx >= BB * HH) return;
    const int b = idx >> 11;
    const int n = idx & (HH - 1);
    const float* __restrict__ g = gates + (size_t)b * (4 * HH);
    const float gi = g[n];
    const float gf = g[HH + n];
    const float gg = g[2 * HH + n];
    const float go = g[3 * HH + n];
    const float c  = sigf(gf) * c_prev[idx] + sigf(gi) * tanhf(gg);
    c_out[idx] = c;
    h_out[idx] = sigf(go) * tanhf(c);
}

// ---------------------------------------------------------------------------
extern "C" void kernel_launch(void* const* d_in, const int* in_sizes, int n_in,
                              void* d_out, int out_size, void* d_ws, size_t ws_size,
                              hipStream_t stream)
{
    (void)in_sizes; (void)n_in; (void)out_size; (void)ws_size;
    const float* x      = (const float*)d_in[0];
    const float* h0     = (const float*)d_in[1];
    const float* c0     = (const float*)d_in[2];
    const float* enc    = (const float*)d_in[3];
    const float* attn_W = (const float*)d_in[4];
    const float* attn_b = (const float*)d_in[5];
    const float* comb_W = (const float*)d_in[6];
    const float* comb_b = (const float*)d_in[7];
    const float* Wih0   = (const float*)d_in[8];
    const float* Whh0   = (const float*)d_in[9];
    const float* bih0   = (const float*)d_in[10];
    const float* bhh0   = (const float*)d_in[11];
    const float* Wih1   = (const float*)d_in[12];
    const float* Whh1   = (const float*)d_in[13];
    const float* bih1   = (const float*)d_in[14];
    const float* bhh1   = (const float*)d_in[15];
    const float* out_W  = (const float*)d_in[16];
    const float* out_b  = (const float*)d_in[17];

    // Workspace carve-up (floats): xc[B,3072] | xin[B,1024] | gates[B,8192]
    float* ws    = (float*)d_ws;
    float* xc    = ws;                         // 786432
    float* xin   = xc + (size_t)BB * KCAT;     // 262144
    float* gates = xin + (size_t)BB * OO;      // 2097152

    // Output carve-up: pred[1,B,O] | h_new[2,B,H2] | c_new[2,B,H2]
    float* pred  = (float*)d_out;
    float* h_new = pred + (size_t)BB * OO;
    float* c_new = h_new + (size_t)2 * BB * HH;
    const size_t layer_off = (size_t)BB * HH;  // 524288

    // 1) attention -> xc = [x0 | attn_applied]
    attn_kernel<<<BB, 256, 0, stream>>>(x, h0, enc, attn_W, attn_b, xc);

    // 2) xin = relu(xc @ comb_W^T + comb_b)          M=256 N=1024 K=3072
    wmma_gemm_kernel<<<dim3(OO / 128, BB / 64), 256, 0, stream>>>(
        xc, comb_W, comb_b, nullptr, xin, BB, OO, KCAT, /*relu*/1);

    // 3) layer-0 gates: xin@Wih0^T + bih0 + bhh0, then += h0[0]@Whh0^T
    wmma_gemm_kernel<<<dim3((4 * HH) / 128, BB / 64), 256, 0, stream>>>(
        xin, Wih0, bih0, bhh0, gates, BB, 4 * HH, OO, 0);
    wmma_gemm_kernel<<<dim3((4 * HH) / 128, BB / 64), 256, 0, stream>>>(
        h0, Whh0, nullptr, nullptr, gates, BB, 4 * HH, HH, /*acc*/2);
    lstm_cell_kernel<<<(BB * HH) / 256, 256, 0, stream>>>(gates, c0, h_new, c_new);

    // 4) layer-1 gates: h1@Wih1^T + bih1 + bhh1, then += h0[1]@Whh1^T
    wmma_gemm_kernel<<<dim3((4 * HH) / 128, BB / 64), 256, 0, stream>>>(
        h_new, Wih1, bih1, bhh1, gates, BB, 4 * HH, HH, 0);
    wmma_gemm_kernel<<<dim3((4 * HH) / 128, BB / 64), 256, 0, stream>>>(
        h0 + layer_off, Whh1, nullptr, nullptr, gates, BB, 4 * HH, HH, /*acc*/2);
    lstm_cell_kernel<<<(BB * HH) / 256, 256, 0, stream>>>(
        gates, c0 + layer_off, h_new + layer_off, c_new + layer_off);

    // 5) prediction = h2 @ out_W^T + out_b           M=256 N=1024 K=2048
    wmma_gemm_kernel<<<dim3(OO / 128, BB / 64), 256, 0, stream>>>(
        h_new + layer_off, out_W, out_b, nullptr, pred, BB, OO, HH, 0);
}